// InfoNCE_10453950398500
// MI455X (gfx1250) — compile-verified
//
#include <hip/hip_runtime.h>
#include <hip/hip_bf16.h>
#include <math.h>

typedef __attribute__((ext_vector_type(2))) float v2f;
typedef __attribute__((ext_vector_type(8))) float v8f;
typedef int v4i __attribute__((vector_size(16)));
typedef __attribute__((address_space(1))) v4i* gas_v4i_ptr;
typedef __attribute__((address_space(3))) v4i* las_v4i_ptr;

#define N_SAMP 1024
#define XDIM   128
#define HID    256
#define PADK   260      // 256 + 4 floats: 16B skew/row -> conflict-free LDS
#define NCHUNK 4        // j-chunks of 256 columns

// Async global -> LDS copy (b128), gfx1250. ASYNCcnt-tracked, bypasses VGPRs.
__device__ __forceinline__ void async_copy_b128(const float* gsrc, float* ldst) {
    __builtin_amdgcn_global_load_async_to_lds_b128(
        (gas_v4i_ptr)gsrc, (las_v4i_ptr)ldst, 0, 0);
}
__device__ __forceinline__ void wait_async0() {
    asm volatile("s_wait_asynccnt 0x0" ::: "memory");
}

// ---------------------------------------------------------------------------
// Kernel 1: fp32 WMMA GEMMs, 16x128 strip per wave (8 accumulators) so one
// A fragment feeds 8 v_wmma_f32_16x16x4_f32 per k-step.
//   strip < 128 : xp = x @ W1[0:128]            (1024x256)
//   strip >= 128: a  = y @ W1[128:256] + b1     (1024x256)
// ---------------------------------------------------------------------------
__global__ __launch_bounds__(256) void infonce_gemm16(
    const float* __restrict__ X, const float* __restrict__ Y,
    const float* __restrict__ W1, const float* __restrict__ b1,
    float* __restrict__ xp, float* __restrict__ aY)
{
    const int lane  = threadIdx.x & 31;
    const int wave  = threadIdx.x >> 5;
    int strip = blockIdx.x * 8 + wave;           // 0..255
    const bool isY = strip >= 128;
    const int  s   = isY ? (strip - 128) : strip;
    const int  mt  = s >> 1;                     // 0..63 row tile
    const int  hh  = s & 1;                      // which 128-col half

    const float* A = isY ? Y : X;
    const float* W = W1 + (isY ? (XDIM * HID) : 0);
    float* OUT     = isY ? aY : xp;

    const int m0    = mt * 16;
    const int half  = lane >> 4;                 // 0: K {0,1}; 1: K {2,3}
    const int l     = lane & 15;
    const int mA    = m0 + l;                    // A row this lane supplies
    const int nbase = hh * 128 + l;

    v8f c[8] = {};
    for (int kb = 0; kb < XDIM; kb += 4) {
        const int k = kb + half * 2;
        v2f a;
        a.x = A[mA * XDIM + k];
        a.y = A[mA * XDIM + k + 1];
        const float* wr0 = &W[(k    ) * HID + nbase];
        const float* wr1 = &W[(k + 1) * HID + nbase];
#pragma unroll
        for (int nt = 0; nt < 8; ++nt) {
            v2f b;
            b.x = wr0[nt * 16];
            b.y = wr1[nt * 16];
            c[nt] = __builtin_amdgcn_wmma_f32_16x16x4_f32(
                        false, a, false, b, (short)0, c[nt], false, false);
        }
    }
#pragma unroll
    for (int nt = 0; nt < 8; ++nt) {
        const int nB = hh * 128 + nt * 16 + l;
        const float bias = isY ? b1[nB] : 0.0f;
#pragma unroll
        for (int r = 0; r < 8; ++r)
            OUT[(m0 + r + half * 8) * HID + nB] = c[nt][r] + bias;
    }
}

// ---------------------------------------------------------------------------
// Kernel 2: pairwise T1 scores + online logsumexp partials.
// Block = 256 threads; blockIdx.x = i-block (16 rows), blockIdx.y = j-chunk
// (256 cols). xp sub-tiles (32 rows) double-buffered via async global->LDS.
// ---------------------------------------------------------------------------
__global__ __launch_bounds__(256) void infonce_pairwise(
    const float* __restrict__ aY,   // 1024x256  (yp + b1)
    const float* __restrict__ xp,   // 1024x256
    const float* __restrict__ W2,   // 256
    float* __restrict__ partM,      // 1024 x NCHUNK
    float* __restrict__ partS,      // 1024 x NCHUNK
    float* __restrict__ diag)       // 1024   (T1[i,i] without b2)
{
    __shared__ __align__(16) float sA[16 * PADK];
    __shared__ __align__(16) float sX[2][32 * PADK];
    __shared__ __align__(16) float sW[HID];
    __shared__ float redM[256];
    __shared__ float redS[256];

    const int t     = threadIdx.x;
    const int ib    = blockIdx.x;     // 0..63
    const int chunk = blockIdx.y;     // 0..3
    const int jbase = chunk * 256;

    sW[t] = W2[t];
#pragma unroll
    for (int q = 0; q < 4; ++q) {     // a-tile: 16 x 256 floats, async
        const int fi  = t + 256 * q;  // float4 index
        const int row = fi >> 6, c4 = fi & 63;
        async_copy_b128(&aY[(ib * 16 + row) * HID + c4 * 4], &sA[row * PADK + c4 * 4]);
    }
#pragma unroll
    for (int q = 0; q < 8; ++q) {     // first 32 xp rows into buffer 0
        const int fi  = t + 256 * q;
        const int row = fi >> 6, c4 = fi & 63;
        async_copy_b128(&xp[(jbase + row) * HID + c4 * 4], &sX[0][row * PADK + c4 * 4]);
    }
    wait_async0();
    __syncthreads();

    const int il    = t & 15;
    const int jslot = t >> 4;
    float mt = -3.0e38f;
    float st = 0.0f;

    for (int sub = 0; sub < 8; ++sub) {
        const int j0 = jbase + sub * 32;
        const float* cur = sX[sub & 1];
        if (sub < 7) {                // prefetch next sub-tile into other buffer
            float* nxt = sX[(sub + 1) & 1];
            const int jn = j0 + 32;
#pragma unroll
            for (int q = 0; q < 8; ++q) {
                const int fi  = t + 256 * q;
                const int row = fi >> 6, c4 = fi & 63;
                async_copy_b128(&xp[(jn + row) * HID + c4 * 4], &nxt[row * PADK + c4 * 4]);
            }
        }

        float acc0 = 0.0f, acc1 = 0.0f;
        const float* ar = &sA[il * PADK];
        const float* x0 = &cur[jslot * PADK];
        const float* x1 = &cur[(jslot + 16) * PADK];
#pragma unroll 4
        for (int k4 = 0; k4 < 64; ++k4) {
            const float4 av = *(const float4*)&ar[k4 * 4];
            const float4 wv = *(const float4*)&sW[k4 * 4];
            const float4 u0 = *(const float4*)&x0[k4 * 4];
            const float4 u1 = *(const float4*)&x1[k4 * 4];
            acc0 = fmaf(fmaxf(av.x + u0.x, 0.0f), wv.x, acc0);
            acc0 = fmaf(fmaxf(av.y + u0.y, 0.0f), wv.y, acc0);
            acc0 = fmaf(fmaxf(av.z + u0.z, 0.0f), wv.z, acc0);
            acc0 = fmaf(fmaxf(av.w + u0.w, 0.0f), wv.w, acc0);
            acc1 = fmaf(fmaxf(av.x + u1.x, 0.0f), wv.x, acc1);
            acc1 = fmaf(fmaxf(av.y + u1.y, 0.0f), wv.y, acc1);
            acc1 = fmaf(fmaxf(av.z + u1.z, 0.0f), wv.z, acc1);
            acc1 = fmaf(fmaxf(av.w + u1.w, 0.0f), wv.w, acc1);
        }

        // online logsumexp update over this thread's j values
        float nm = fmaxf(mt, acc0);
        st = st * __expf(mt - nm) + __expf(acc0 - nm);
        mt = nm;
        nm = fmaxf(mt, acc1);
        st = st * __expf(mt - nm) + __expf(acc1 - nm);
        mt = nm;

        const int ig = ib * 16 + il;
        if (ig == j0 + jslot)      diag[ig] = acc0;   // T0[i] == T1[i,i]
        if (ig == j0 + jslot + 16) diag[ig] = acc1;

        if (sub < 7) {
            wait_async0();            // my prefetch landed
            __syncthreads();          // everyone's prefetch visible; buf released
        }
    }

    redM[t] = mt;
    redS[t] = st;
    __syncthreads();
    if (t < 16) {                     // merge the 16 threads sharing row i
        float M = -3.0e38f;
#pragma unroll
        for (int q = 0; q < 16; ++q) M = fmaxf(M, redM[t + q * 16]);
        float S = 0.0f;
#pragma unroll
        for (int q = 0; q < 16; ++q) S += redS[t + q * 16] * __expf(redM[t + q * 16] - M);
        const int ig = ib * 16 + t;
        partM[ig * NCHUNK + chunk] = M;
        partS[ig * NCHUNK + chunk] = S;
    }
}

// ---------------------------------------------------------------------------
// Kernel 3: merge chunk partials -> lse[i]; output mean(diag - lse) + log N.
// (b2 cancels exactly between T0 and lse.)
// ---------------------------------------------------------------------------
__global__ __launch_bounds__(1024) void infonce_finalize(
    const float* __restrict__ partM, const float* __restrict__ partS,
    const float* __restrict__ diag, float* __restrict__ out)
{
    __shared__ float red[1024];
    const int i = threadIdx.x;
    float M = -3.0e38f;
#pragma unroll
    for (int c = 0; c < NCHUNK; ++c) M = fmaxf(M, partM[i * NCHUNK + c]);
    float S = 0.0f;
#pragma unroll
    for (int c = 0; c < NCHUNK; ++c) S += partS[i * NCHUNK + c] * __expf(partM[i * NCHUNK + c] - M);
    const float lse = M + __logf(S);
    red[i] = diag[i] - lse;
    __syncthreads();
    for (int off = 512; off > 0; off >>= 1) {
        if (i < off) red[i] += red[i + off];
        __syncthreads();
    }
    if (i == 0) out[0] = red[0] / (float)N_SAMP + __logf((float)N_SAMP);
}

// ---------------------------------------------------------------------------
extern "C" void kernel_launch(void* const* d_in, const int* in_sizes, int n_in,
                              void* d_out, int out_size, void* d_ws, size_t ws_size,
                              hipStream_t stream) {
    const float* x  = (const float*)d_in[0];   // 1024x128
    const float* y  = (const float*)d_in[1];   // 1024x128
    const float* W1 = (const float*)d_in[2];   // 256x256
    const float* b1 = (const float*)d_in[3];   // 256
    const float* W2 = (const float*)d_in[4];   // 256x1
    float* out = (float*)d_out;

    float* ws    = (float*)d_ws;
    float* xp    = ws;                             // 1024*256
    float* aY    = ws + N_SAMP * HID;              // 1024*256
    float* partM = ws + 2 * N_SAMP * HID;          // 1024*4
    float* partS = partM + N_SAMP * NCHUNK;        // 1024*4
    float* diag  = partS + N_SAMP * NCHUNK;        // 1024

    infonce_gemm16<<<32, 256, 0, stream>>>(x, y, W1, b1, xp, aY);
    infonce_pairwise<<<dim3(64, NCHUNK), 256, 0, stream>>>(aY, xp, W2, partM, partS, diag);
    infonce_finalize<<<1, 1024, 0, stream>>>(partM, partS, diag, out);
}